// MixedStateTreeGenerator_73727408603448
// MI455X (gfx1250) — compile-verified
//
#include <hip/hip_runtime.h>
#include <math.h>

// ---------------------------------------------------------------------------
// MixedStateTreeGenerator on MI455X (gfx1250)
//
// Decomposition:
//  K1 dfs_structure_kernel : replay the (data-independent) DFS stack machine
//                            -> parent/obs/len per tree slot, level buckets,
//                            size scalar.
//  K2 init_kernel          : expT = exp(ltm) (the GEMM A-matrix, 64x32 f32),
//                            expleft, root state in rescaled-exp form.
//  K3 level_kernel (d=1..32): batched log-semiring matvec as a real f32 GEMM
//                            Y(64 x N_d) = expT(64x32) * E(32 x N_d)
//                            using V_WMMA_F32_16X16X4_F32 (32 WMMAs / tile),
//                            then per-node: m' = m + log(maxY),
//                            E' = Y/maxY, prob = m' + log(sum expleft*E').
//  K4 seq_kernel           : materialize sequences + lengths by parent walk.
// ---------------------------------------------------------------------------

#define MAX_TREE 16384
#define MAX_LEN 32
#define NUM_OBS 2
#define NS 32

typedef __attribute__((ext_vector_type(2))) float v2f;
typedef __attribute__((ext_vector_type(8))) float v8f;

// ---------------- K1: structural DFS replay (pure int, deterministic) ------
__global__ void dfs_structure_kernel(int* __restrict__ parent,
                                     int* __restrict__ obsb,
                                     int* __restrict__ lenA,
                                     int* __restrict__ levelNodes,
                                     int* __restrict__ levelStart,
                                     int* __restrict__ sizeOut) {
  if (threadIdx.x != 0 || blockIdx.x != 0) return;
  // DFS stack: max depth ~34 (one sibling left behind per level)
  int stk_p[80], stk_o[80], stk_l[80];
  int sp = 0;
  stk_p[sp] = -1; stk_o[sp] = -1; stk_l[sp] = 0; sp = 1;  // root
  int tsize = 0;
  while (sp > 0 && tsize < MAX_TREE) {
    sp--;
    int p = stk_p[sp], o = stk_o[sp], l = stk_l[sp];
    int t = tsize++;
    parent[t] = p; obsb[t] = o; lenA[t] = l;
    if (l < MAX_LEN) {
      // reference pushes o=0 then o=1 -> o=1 popped first (matches .at order)
      stk_p[sp] = t; stk_o[sp] = 0; stk_l[sp] = l + 1; sp++;
      stk_p[sp] = t; stk_o[sp] = 1; stk_l[sp] = l + 1; sp++;
    }
  }
  *sizeOut = tsize;  // == MAX_TREE
  // bucket nodes by depth (33 levels)
  int cnt[MAX_LEN + 1];
  for (int d = 0; d <= MAX_LEN; ++d) cnt[d] = 0;
  for (int t = 0; t < tsize; ++t) cnt[lenA[t]]++;
  int run = 0;
  for (int d = 0; d <= MAX_LEN; ++d) { levelStart[d] = run; run += cnt[d]; cnt[d] = 0; }
  levelStart[MAX_LEN + 1] = run;
  for (int t = 0; t < tsize; ++t) {
    int d = lenA[t];
    levelNodes[levelStart[d] + cnt[d]++] = t;
  }
}

// ---------------- K2: constants + root state -------------------------------
__global__ void init_kernel(const float* __restrict__ ltm,
                            const float* __restrict__ left,
                            const float* __restrict__ right,
                            float* __restrict__ expT,
                            float* __restrict__ expleft,
                            float* __restrict__ mArr,
                            float* __restrict__ expstate,
                            float* __restrict__ probOut) {
  int tid = blockIdx.x * blockDim.x + threadIdx.x;
  if (tid < NUM_OBS * NS * NS) expT[tid] = expf(ltm[tid]);  // row = o*32+i, col = j
  if (tid < NS) expleft[tid] = expf(left[tid]);
  if (tid == 0) {
    float m0 = right[0];
    for (int j = 1; j < NS; ++j) m0 = fmaxf(m0, right[j]);
    float acc = 0.f;
    for (int j = 0; j < NS; ++j) {
      float es = expf(right[j] - m0);
      expstate[j] = es;                     // node 0 == root (first popped)
      acc += expf(left[j] + right[j] - m0);
    }
    mArr[0] = m0;
    probOut[0] = m0 + logf(acc);            // logsumexp(left + right)
  }
}

// ---------------- K3: one depth level, WMMA f32 GEMM -----------------------
// Block = 1 wave (32 lanes), handles a tile of 16 nodes (GEMM columns).
// Y(64 x 16) = expT(64 x 32) * E(32 x 16): 4 M-tiles x 8 K-steps of 16x16x4.
__global__ __launch_bounds__(32)
void level_kernel(int d,
                  const int* __restrict__ levelStart,
                  const int* __restrict__ levelNodes,
                  const int* __restrict__ parent,
                  const int* __restrict__ obsb,
                  const float* __restrict__ expT,
                  const float* __restrict__ expleft,
                  float* __restrict__ mArr,
                  float* __restrict__ expstate,
                  float* __restrict__ probOut) {
  const int start = levelStart[d];
  const int count = levelStart[d + 1] - start;
  const int base = blockIdx.x * 16;
  if (base >= count) return;           // uniform early-exit (over-launched grid)

  const int lane = threadIdx.x;        // 0..31
  const int mrow = lane & 15;
  const int khalf = (lane < 16) ? 0 : 2;

  __shared__ float Bcol[NS][17];       // E columns: Bcol[j][c], padded
  __shared__ float Y[64][17];          // GEMM result staging
  __shared__ int colT[16], colP[16], colO[16];

  if (lane < 16) {
    int c = base + lane;
    int t = (c < count) ? levelNodes[start + c] : levelNodes[start];  // dup col pad
    colT[lane] = (c < count) ? t : -1;
    colP[lane] = parent[t];
    colO[lane] = obsb[t];
  }
  __syncthreads();

  // Stage B: column c = exp-rescaled parent state (32 f32, contiguous)
  if (lane < 16) {
    const float* src = expstate + (size_t)colP[lane] * NS;
    for (int j = 0; j < NS; ++j) Bcol[j][lane] = src[j];
  }
  __syncthreads();

  // Preload A fragments (ISA 7.12.2, 32-bit A 16x4):
  //   lanes 0-15: M=lane, {K0,K1}; lanes 16-31: M=lane-16, {K2,K3}
  v2f afr[4][8];
  for (int m = 0; m < 4; ++m)
    for (int k = 0; k < 8; ++k) {
      int row = m * 16 + mrow;
      int col = k * 4 + khalf;
      v2f a; a.x = expT[row * NS + col]; a.y = expT[row * NS + col + 1];
      afr[m][k] = a;
    }

  v8f acc[4] = {};
  for (int k = 0; k < 8; ++k) {
    // B fragment (4x16): N = lane&15; VGPR0 holds {K0|K2}, VGPR1 holds {K1|K3}
    v2f b;
    b.x = Bcol[k * 4 + khalf][mrow];
    b.y = Bcol[k * 4 + khalf + 1][mrow];
    for (int m = 0; m < 4; ++m) {
      acc[m] = __builtin_amdgcn_wmma_f32_16x16x4_f32(
          /*neg_a=*/false, afr[m][k], /*neg_b=*/false, b,
          /*c_mod=*/(short)0, acc[m], /*reuse_a=*/false, /*reuse_b=*/false);
    }
  }

  // D layout (32-bit C/D 16x16): VGPR r -> rows r (lanes 0-15) and r+8 (16-31)
  for (int m = 0; m < 4; ++m)
    for (int r = 0; r < 8; ++r) {
      int row = m * 16 + ((lane < 16) ? r : r + 8);
      Y[row][mrow] = acc[m][r];
    }
  __syncthreads();

  // Epilogue: per node, rescale and emit prob.  state' = log(Y)+m_parent;
  // m' = m_parent + log(maxY); E' = Y/maxY; prob = m' + log(sum expleft*E')
  if (lane < 16 && colT[lane] >= 0) {
    int t = colT[lane], o = colO[lane], p = colP[lane];
    float mx = Y[o * NS][lane];
    for (int i = 1; i < NS; ++i) mx = fmaxf(mx, Y[o * NS + i][lane]);
    float mprime = mArr[p] + logf(mx);
    float inv = 1.0f / mx;
    float s = 0.f;
    float* dst = expstate + (size_t)t * NS;
    for (int i = 0; i < NS; ++i) {
      float es = Y[o * NS + i][lane] * inv;
      dst[i] = es;
      s += expleft[i] * es;
    }
    mArr[t] = mprime;
    probOut[t] = mprime + logf(s);
  }
}

// ---------------- K4: sequences + lengths ----------------------------------
__global__ void seq_kernel(const int* __restrict__ parent,
                           const int* __restrict__ obsb,
                           const int* __restrict__ lenA,
                           int* __restrict__ outSeq,
                           int* __restrict__ outLen) {
  int t = blockIdx.x * blockDim.x + threadIdx.x;
  if (t >= MAX_TREE) return;
  int L = lenA[t];
  outLen[t] = L;
  int* row = outSeq + (size_t)t * MAX_LEN;
  for (int k = L; k < MAX_LEN; ++k) row[k] = 0;   // inherited zeros
  int cur = t;
  while (cur > 0) {                                // stop at root
    row[lenA[cur] - 1] = obsb[cur];
    cur = parent[cur];
  }
}

// ---------------------------------------------------------------------------
extern "C" void kernel_launch(void* const* d_in, const int* in_sizes, int n_in,
                              void* d_out, int out_size, void* d_ws, size_t ws_size,
                              hipStream_t stream) {
  (void)in_sizes; (void)n_in; (void)out_size; (void)ws_size;
  const float* ltm   = (const float*)d_in[0];   // [2,32,32]
  const float* left  = (const float*)d_in[1];   // [32]
  const float* right = (const float*)d_in[2];   // [32]

  // d_out: seq[16384*32] i32 | len[16384] i32 | prob[16384] f32 | size i32
  int*   outSeq  = (int*)d_out;
  int*   outLen  = outSeq + MAX_TREE * MAX_LEN;
  float* probOut = (float*)(outSeq + MAX_TREE * (MAX_LEN + 1));
  int*   sizeOut = outSeq + MAX_TREE * (MAX_LEN + 2);

  // workspace carve-up (~2.4 MB)
  int* parent     = (int*)d_ws;
  int* obsb       = parent + MAX_TREE;
  int* lenA       = obsb + MAX_TREE;
  int* levelNodes = lenA + MAX_TREE;
  int* levelStart = levelNodes + MAX_TREE;              // 34 ints (pad to 64)
  float* expT     = (float*)(levelStart + 64);          // 64x32
  float* expleft  = expT + NUM_OBS * NS * NS;           // 32
  float* mArr     = expleft + NS;                       // 16384
  float* expstate = mArr + MAX_TREE;                    // 16384x32

  dfs_structure_kernel<<<1, 1, 0, stream>>>(parent, obsb, lenA, levelNodes,
                                            levelStart, sizeOut);
  init_kernel<<<8, 256, 0, stream>>>(ltm, left, right, expT, expleft, mArr,
                                     expstate, probOut);
  const int LEVEL_BLOCKS = MAX_TREE / 16;  // over-launch; blocks bound-check
  for (int d = 1; d <= MAX_LEN; ++d) {
    level_kernel<<<LEVEL_BLOCKS, 32, 0, stream>>>(d, levelStart, levelNodes,
                                                  parent, obsb, expT, expleft,
                                                  mArr, expstate, probOut);
  }
  seq_kernel<<<MAX_TREE / 256, 256, 0, stream>>>(parent, obsb, lenA, outSeq,
                                                 outLen);
}